// ConstraintScreeningGNN_12317966205314
// MI455X (gfx1250) — compile-verified
//
#include <hip/hip_runtime.h>
#include <hip/hip_bf16.h>

typedef __attribute__((ext_vector_type(2))) float v2f;
typedef __attribute__((ext_vector_type(8))) float v8f;

#define LN_EPS 1e-5f

__device__ __forceinline__ float leakyf(float x, float s) { return x >= 0.f ? x : s * x; }

// ---------------- embed: h = leaky(LN(x @ W + b), 0.1);  x:[N,16], W:[16,64]
__global__ void k_embed(const float* __restrict__ x, const float* __restrict__ w,
                        const float* __restrict__ b, const float* __restrict__ g,
                        const float* __restrict__ beta, float* __restrict__ h, int N) {
  __shared__ float xs[16];
  __shared__ float sbuf[64];
  int n = blockIdx.x;
  int c = threadIdx.x;  // 0..63
  if (c < 16) xs[c] = x[(size_t)n * 16 + c];
  __syncthreads();
  float acc = b[c];
#pragma unroll
  for (int k = 0; k < 16; ++k) acc = fmaf(xs[k], w[k * 64 + c], acc);
  sbuf[c] = acc;
  __syncthreads();
  for (int s = 32; s > 0; s >>= 1) { if (c < s) sbuf[c] += sbuf[c + s]; __syncthreads(); }
  float mean = sbuf[0] * (1.f / 64.f);
  __syncthreads();
  float d = acc - mean;
  sbuf[c] = d * d;
  __syncthreads();
  for (int s = 32; s > 0; s >>= 1) { if (c < s) sbuf[c] += sbuf[c + s]; __syncthreads(); }
  float var = sbuf[0] * (1.f / 64.f);
  float y = d * rsqrtf(var + LN_EPS) * g[c] + beta[c];
  h[(size_t)n * 64 + c] = leakyf(y, 0.1f);
}

// ---------------- hp = h @ W via V_WMMA_F32_16X16X4_F32.  h:[M,64], W:[64,128], hp:[M,128]
// One wave per 16x16 output tile, K=64 -> 16 wmma steps.
__global__ void k_gemm_wmma(const float* __restrict__ A, const float* __restrict__ B,
                            float* __restrict__ C, int M) {
  int waveGlobal = (int)((blockIdx.x * blockDim.x + threadIdx.x) >> 5);
  int lane = threadIdx.x & 31;
  int rowTile = waveGlobal >> 3;   // 8 column tiles (128/16)
  int colTile = waveGlobal & 7;
  int row0 = rowTile * 16;
  if (row0 >= M) return;           // wave-uniform
  int col0 = colTile * 16;
  int r = lane & 15;
  int hi = lane >> 4;
  int arow = row0 + r;
  if (arow >= M) arow = M - 1;     // clamp: keep EXEC all-ones for WMMA
  const float* __restrict__ ap = A + (size_t)arow * 64;
  v8f acc = {};
#pragma unroll
  for (int kk = 0; kk < 16; ++kk) {
    int k0 = kk * 4 + 2 * hi;
    v2f a, b;
    a.x = ap[k0];
    a.y = ap[k0 + 1];
    b.x = B[k0 * 128 + col0 + r];
    b.y = B[(k0 + 1) * 128 + col0 + r];
    acc = __builtin_amdgcn_wmma_f32_16x16x4_f32(false, a, false, b, (short)0, acc,
                                                false, false);
  }
#pragma unroll
  for (int v = 0; v < 8; ++v) {
    int row = row0 + v + 8 * hi;
    if (row < M) C[(size_t)row * 128 + col0 + r] = acc[v];
  }
}

// ---------------- attention logits: al_src/al_dst [N,2]
__global__ void k_att(const float* __restrict__ hp, const float* __restrict__ a_src,
                      const float* __restrict__ a_dst, float* __restrict__ al_s,
                      float* __restrict__ al_d, int N) {
  int idx = blockIdx.x * blockDim.x + threadIdx.x;
  if (idx >= N * 2) return;
  int n = idx >> 1, hd = idx & 1;
  const float* __restrict__ row = hp + (size_t)n * 128 + hd * 64;
  const float* __restrict__ as = a_src + hd * 64;
  const float* __restrict__ ad = a_dst + hd * 64;
  float s0 = 0.f, s1 = 0.f;
#pragma unroll 8
  for (int c = 0; c < 64; ++c) {
    float v = row[c];
    s0 = fmaf(v, as[c], s0);
    s1 = fmaf(v, ad[c], s1);
  }
  al_s[idx] = s0;
  al_d[idx] = s1;
}

// ---------------- per-layer scratch clear: accum=0, denom=0, emax_enc=0
__global__ void k_clear(float* __restrict__ accum, float* __restrict__ denom,
                        unsigned* __restrict__ emax, int N) {
  long i = (long)blockIdx.x * blockDim.x + threadIdx.x;
  long na = (long)N * 128;
  if (i < na) accum[i] = 0.f;
  else if (i < na + 2L * N) denom[i - na] = 0.f;
  else if (i < na + 4L * N) emax[i - na - 2L * N] = 0u;  // enc(-inf)-safe floor
}

// order-preserving float->uint encoding
__device__ __forceinline__ unsigned enc_f32(float f) {
  unsigned u = __float_as_uint(f);
  return u ^ ((u >> 31) ? 0xFFFFFFFFu : 0x80000000u);
}
__device__ __forceinline__ float dec_f32(unsigned u) {
  u ^= ((u >> 31) ? 0x80000000u : 0xFFFFFFFFu);
  return __uint_as_float(u);
}

// ---------------- pass A: segment max of leaky attention logits (thread per edge*head)
__global__ void k_edge_max(const int* __restrict__ ei, const float* __restrict__ al_s,
                           const float* __restrict__ al_d, unsigned* __restrict__ emax,
                           int E, int N) {
  long idx = (long)blockIdx.x * blockDim.x + threadIdx.x;
  if (idx >= ((long)E + N) * 2) return;
  long e = idx >> 1;
  int hd = (int)(idx & 1);
  int s, d;
  if (e < E) { s = ei[e]; d = ei[(long)E + e]; }
  else       { s = d = (int)(e - E); }  // self-loops appended
  float v = al_s[(size_t)s * 2 + hd] + al_d[(size_t)d * 2 + hd];
  v = leakyf(v, 0.2f);
  atomicMax(&emax[(size_t)d * 2 + hd], enc_f32(v));
}

// ---------------- pass B: accum[dst] += ee * hp[src]; denom[dst] += ee  (wave per edge)
__global__ void k_edge_msg(const int* __restrict__ ei, const float* __restrict__ al_s,
                           const float* __restrict__ al_d, const unsigned* __restrict__ emax,
                           const float* __restrict__ hp, float* __restrict__ denom,
                           float* __restrict__ accum, int E, int N) {
  long gid = (long)blockIdx.x * blockDim.x + threadIdx.x;
  long e = gid >> 5;
  int lane = (int)(gid & 31);
  if (e >= (long)E + N) return;  // wave-uniform
  int s, d;
  if (e < E) { s = ei[e]; d = ei[(long)E + e]; }
  else       { s = d = (int)(e - E); }
  int hd = lane >> 4;  // lanes 0-15: head0 (cols 0-63), 16-31: head1 (cols 64-127)
  float v = al_s[(size_t)s * 2 + hd] + al_d[(size_t)d * 2 + hd];
  v = leakyf(v, 0.2f);
  float m = dec_f32(emax[(size_t)d * 2 + hd]);
  float ee = __expf(v - m);
  if ((lane & 15) == 0) atomicAdd(&denom[(size_t)d * 2 + hd], ee);
  const float4 hv = *((const float4*)(hp + (size_t)s * 128) + lane);
  float* __restrict__ ab = accum + (size_t)d * 128 + lane * 4;
  atomicAdd(ab + 0, ee * hv.x);
  atomicAdd(ab + 1, ee * hv.y);
  atomicAdd(ab + 2, ee * hv.z);
  atomicAdd(ab + 3, ee * hv.w);
}

// ---------------- node finalize: h = leaky(LN(mean_heads(accum/denom) + conv_b), 0.1)
__global__ void k_finalize(const float* __restrict__ accum, const float* __restrict__ denom,
                           const float* __restrict__ cb, const float* __restrict__ g,
                           const float* __restrict__ b, float* __restrict__ h, int N) {
  __shared__ float sbuf[64];
  int n = blockIdx.x, c = threadIdx.x;
  float d0 = denom[(size_t)n * 2 + 0] + 1e-16f;
  float d1 = denom[(size_t)n * 2 + 1] + 1e-16f;
  float v = 0.5f * (accum[(size_t)n * 128 + c] / d0 +
                    accum[(size_t)n * 128 + 64 + c] / d1) + cb[c];
  sbuf[c] = v;
  __syncthreads();
  for (int s = 32; s > 0; s >>= 1) { if (c < s) sbuf[c] += sbuf[c + s]; __syncthreads(); }
  float mean = sbuf[0] * (1.f / 64.f);
  __syncthreads();
  float dd = v - mean;
  sbuf[c] = dd * dd;
  __syncthreads();
  for (int s = 32; s > 0; s >>= 1) { if (c < s) sbuf[c] += sbuf[c + s]; __syncthreads(); }
  float var = sbuf[0] * (1.f / 64.f);
  float y = dd * rsqrtf(var + LN_EPS) * g[c] + b[c];
  h[(size_t)n * 64 + c] = leakyf(y, 0.1f);
}

// ---------------- global mean pool (sum; divided in head)
__global__ void k_zero_pool(float* __restrict__ pool) { pool[threadIdx.x] = 0.f; }

__global__ void k_pool(const float* __restrict__ h, float* __restrict__ pool, int N) {
  int c = threadIdx.x;  // 64
  float s = 0.f;
  for (int n = blockIdx.x; n < N; n += gridDim.x) s += h[(size_t)n * 64 + c];
  atomicAdd(&pool[c], s);
}

// ---------------- MLP head: sigmoid(leaky(LN(g@W1+b1))@W2 + b2)
__global__ void k_head(const float* __restrict__ pool, const float* __restrict__ w1,
                       const float* __restrict__ b1, const float* __restrict__ g,
                       const float* __restrict__ bb, const float* __restrict__ w2,
                       const float* __restrict__ b2, float* __restrict__ out,
                       int N, int OUT) {
  __shared__ float gv[64];
  __shared__ float a[64];
  __shared__ float sbuf[64];
  int t = threadIdx.x;  // 256
  if (t < 64) gv[t] = pool[t] / (float)N;
  __syncthreads();
  float acc = 0.f;
  if (t < 64) {
    acc = b1[t];
#pragma unroll
    for (int k = 0; k < 64; ++k) acc = fmaf(gv[k], w1[k * 64 + t], acc);
    sbuf[t] = acc;
  }
  __syncthreads();
  for (int s = 32; s > 0; s >>= 1) { if (t < s) sbuf[t] += sbuf[t + s]; __syncthreads(); }
  float mean = sbuf[0] * (1.f / 64.f);
  __syncthreads();
  float dd = acc - mean;
  if (t < 64) sbuf[t] = dd * dd;
  __syncthreads();
  for (int s = 32; s > 0; s >>= 1) { if (t < s) sbuf[t] += sbuf[t + s]; __syncthreads(); }
  float var = sbuf[0] * (1.f / 64.f);
  if (t < 64) {
    float y = dd * rsqrtf(var + LN_EPS) * g[t] + bb[t];
    a[t] = leakyf(y, 0.1f);
  }
  __syncthreads();
  for (int o = t; o < OUT; o += blockDim.x) {
    float s = b2[o];
#pragma unroll
    for (int k = 0; k < 64; ++k) s = fmaf(a[k], w2[k * OUT + o], s);
    out[o] = 1.f / (1.f + __expf(-s));
  }
}

extern "C" void kernel_launch(void* const* d_in, const int* in_sizes, int n_in,
                              void* d_out, int out_size, void* d_ws, size_t ws_size,
                              hipStream_t stream) {
  const float* x        = (const float*)d_in[0];
  const int*   ei       = (const int*)d_in[1];
  const float* emb_w    = (const float*)d_in[2];
  const float* emb_b    = (const float*)d_in[3];
  const float* emb_g    = (const float*)d_in[4];
  const float* emb_beta = (const float*)d_in[5];
  const float* lin_w    = (const float*)d_in[6];
  const float* att_src  = (const float*)d_in[7];
  const float* att_dst  = (const float*)d_in[8];
  const float* conv_b   = (const float*)d_in[9];
  const float* ln_g     = (const float*)d_in[10];
  const float* ln_b     = (const float*)d_in[11];
  const float* fc1_w    = (const float*)d_in[12];
  const float* fc1_b    = (const float*)d_in[13];
  const float* fcn_g    = (const float*)d_in[14];
  const float* fcn_b    = (const float*)d_in[15];
  const float* fc2_w    = (const float*)d_in[16];
  const float* fc2_b    = (const float*)d_in[17];
  float* out = (float*)d_out;

  const int N = in_sizes[0] / 16;
  const int E = in_sizes[1] / 2;
  const int L = in_sizes[6] / (64 * 128);

  float* ws = (float*)d_ws;
  size_t off = 0;
  float*    h     = ws + off; off += (size_t)N * 64;
  float*    hp    = ws + off; off += (size_t)N * 128;
  float*    accum = ws + off; off += (size_t)N * 128;
  float*    al_s  = ws + off; off += (size_t)N * 2;
  float*    al_d  = ws + off; off += (size_t)N * 2;
  float*    denom = ws + off; off += (size_t)N * 2;
  unsigned* emax  = (unsigned*)(ws + off); off += (size_t)N * 2;
  float*    pool  = ws + off; off += 64;
  (void)ws_size; (void)n_in;

  k_embed<<<N, 64, 0, stream>>>(x, emb_w, emb_b, emb_g, emb_beta, h, N);

  const int rowTiles = (N + 15) / 16;        // 8 waves/block == 8 col tiles -> grid = rowTiles
  const long totE = (long)E + N;
  for (int l = 0; l < L; ++l) {
    k_gemm_wmma<<<rowTiles, 256, 0, stream>>>(h, lin_w + (size_t)l * 64 * 128, hp, N);
    k_att<<<(N * 2 + 255) / 256, 256, 0, stream>>>(hp, att_src + (size_t)l * 128,
                                                   att_dst + (size_t)l * 128, al_s, al_d, N);
    long clrTot = (long)N * 132;
    k_clear<<<(int)((clrTot + 255) / 256), 256, 0, stream>>>(accum, denom, emax, N);
    long pa = totE * 2;
    k_edge_max<<<(int)((pa + 255) / 256), 256, 0, stream>>>(ei, al_s, al_d, emax, E, N);
    long pb = totE * 32;
    k_edge_msg<<<(int)((pb + 255) / 256), 256, 0, stream>>>(ei, al_s, al_d, emax, hp,
                                                            denom, accum, E, N);
    k_finalize<<<N, 64, 0, stream>>>(accum, denom, conv_b + (size_t)l * 64,
                                     ln_g + (size_t)l * 64, ln_b + (size_t)l * 64, h, N);
  }
  k_zero_pool<<<1, 64, 0, stream>>>(pool);
  k_pool<<<512, 64, 0, stream>>>(h, pool, N);
  k_head<<<1, 256, 0, stream>>>(pool, fc1_w, fc1_b, fcn_g, fcn_b, fc2_w, fc2_b,
                                out, N, out_size);
}